// SENGR_GCN_36644660970308
// MI455X (gfx1250) — compile-verified
//
#include <hip/hip_runtime.h>

#define D 64

typedef __attribute__((ext_vector_type(2))) float v2f;
typedef __attribute__((ext_vector_type(8))) float v8f;

// ---------------------------------------------------------------------------
// Zero-fill (float4 stores) -- graph-capture-safe replacement for memset.
// ---------------------------------------------------------------------------
__global__ __launch_bounds__(256) void SENGR_zero_f32(float* __restrict__ p,
                                                      long long n4) {
  long long i = (long long)blockIdx.x * blockDim.x + threadIdx.x;
  if (i < n4) {
    float4 z;
    z.x = 0.f; z.y = 0.f; z.z = 0.f; z.w = 0.f;
    ((float4*)p)[i] = z;
  }
}

// ---------------------------------------------------------------------------
// Edge phase: msg = w * x[src]; agg[dst] += msg   (scatter-add, f32 atomics)
// 16 lanes per edge, each lane handles 4 consecutive features (b128 gather,
// 4x global_atomic_add_f32). x is addressed through a (xa,xb,split) pair so
// layer 1 can read the virtual concat(user_emb, item_emb) without a copy.
// ---------------------------------------------------------------------------
__global__ __launch_bounds__(256) void SENGR_edge_scatter(
    const int* __restrict__ src, const int* __restrict__ dst,
    const float* __restrict__ ew,
    const float* __restrict__ xa, const float* __restrict__ xb, int split,
    float* __restrict__ agg, long long nwork) {
  long long gid = (long long)blockIdx.x * blockDim.x + threadIdx.x;
  if (gid >= nwork) return;
  long long e = gid >> 4;
  int q = (int)(gid & 15);

  int s = src[e];
  int d = dst[e];
  float w = ew[e];

  const float* xr = (s < split) ? (xa + (long long)s * D)
                                : (xb + (long long)(s - split) * D);
  float4 v = *(const float4*)(xr + q * 4);

  float* o = agg + (long long)d * D + q * 4;
  unsafeAtomicAdd(o + 0, w * v.x);
  unsafeAtomicAdd(o + 1, w * v.y);
  unsafeAtomicAdd(o + 2, w * v.z);
  unsafeAtomicAdd(o + 3, w * v.w);
}

// ---------------------------------------------------------------------------
// Linear phase: out = ((agg + x) * 0.5) @ W + bias, via V_WMMA_F32_16X16X4_F32.
// One wave32 per 16-row tile, covering all 64 output columns with 4
// accumulators; A-fragment reused across the 4 column tiles. Bias is folded
// into the accumulator init.
//
// f32 WMMA fragment layouts (ISA 7.12.2):
//   A 16x4 : lane L holds M = L&15; VGPR0/1 = K = 2*(L>>4) + {0,1}
//   B 4x16 : lane L holds N = L&15; VGPR0/1 = K = 2*(L>>4) + {0,1}
//   C 16x16: VGPR r -> M = r + 8*(L>>4), N = L&15
// ---------------------------------------------------------------------------
__global__ __launch_bounds__(256) void SENGR_gcn_linear_wmma(
    const float* __restrict__ agg,
    const float* __restrict__ xa, const float* __restrict__ xb, int split,
    const float* __restrict__ W, const float* __restrict__ bias,
    float* __restrict__ out, int nrows) {
  int wave = threadIdx.x >> 5;
  int lane = threadIdx.x & 31;
  int ntiles = (nrows + 15) >> 4;
  int tile = blockIdx.x * 8 + wave;
  if (tile >= ntiles) return;  // wave-uniform: EXEC stays all-1s for WMMA

  int row0 = tile << 4;
  int half = lane >> 4;   // 0: K+{0,1}   1: K+{2,3}
  int n    = lane & 15;   // A: M index / B,C: N index
  int koff = half << 1;

  // A-row pointers (clamped so tail tiles still keep every lane active)
  int arow = row0 + n;
  if (arow > nrows - 1) arow = nrows - 1;
  const float* xr = (arow < split) ? (xa + (long long)arow * D)
                                   : (xb + (long long)(arow - split) * D);
  const float* gr = agg + (long long)arow * D;

  // Accumulators initialized with the bias (out[i][j] = ... + b[j])
  v8f acc0, acc1, acc2, acc3;
  float bb0 = bias[n], bb1 = bias[16 + n], bb2 = bias[32 + n], bb3 = bias[48 + n];
#pragma unroll
  for (int r = 0; r < 8; ++r) {
    acc0[r] = bb0; acc1[r] = bb1; acc2[r] = bb2; acc3[r] = bb3;
  }

#pragma unroll
  for (int k0 = 0; k0 < D; k0 += 4) {
    int ka = k0 + koff;  // even -> 8B-aligned v2f loads
    v2f g  = *(const v2f*)(gr + ka);
    v2f xv = *(const v2f*)(xr + ka);
    v2f a  = (g + xv) * 0.5f;

    const float* w0 = W + (long long)ka * D;  // row ka   of W
    const float* w1 = w0 + D;                 // row ka+1 of W
    v2f bf;

    bf.x = w0[n];      bf.y = w1[n];
    acc0 = __builtin_amdgcn_wmma_f32_16x16x4_f32(false, a, false, bf, (short)0,
                                                 acc0, false, false);
    bf.x = w0[16 + n]; bf.y = w1[16 + n];
    acc1 = __builtin_amdgcn_wmma_f32_16x16x4_f32(false, a, false, bf, (short)0,
                                                 acc1, false, false);
    bf.x = w0[32 + n]; bf.y = w1[32 + n];
    acc2 = __builtin_amdgcn_wmma_f32_16x16x4_f32(false, a, false, bf, (short)0,
                                                 acc2, false, false);
    bf.x = w0[48 + n]; bf.y = w1[48 + n];
    acc3 = __builtin_amdgcn_wmma_f32_16x16x4_f32(false, a, false, bf, (short)0,
                                                 acc3, false, false);
  }

  // Store C/D: VGPR r -> row row0 + 8*half + r, col = tile*16 + n
#pragma unroll
  for (int r = 0; r < 8; ++r) {
    int grow = row0 + (half << 3) + r;
    if (grow < nrows) {
      long long base = (long long)grow * D + n;
      out[base +  0] = acc0[r];
      out[base + 16] = acc1[r];
      out[base + 32] = acc2[r];
      out[base + 48] = acc3[r];
    }
  }
}

// ---------------------------------------------------------------------------
// Launch: layer1 (concat inputs) -> ws.x1 ; layer2 (x1) -> d_out.
// ws layout: [ agg : N*D floats ][ x1 : N*D floats ]  (~102 MB)
// ---------------------------------------------------------------------------
extern "C" void kernel_launch(void* const* d_in, const int* in_sizes, int n_in,
                              void* d_out, int out_size, void* d_ws,
                              size_t ws_size, hipStream_t stream) {
  const int*   eidx = (const int*)d_in[0];    // [2, E] int32
  const float* ew   = (const float*)d_in[1];  // [E]
  const float* ue   = (const float*)d_in[2];  // [NU, 64]
  const float* ie   = (const float*)d_in[3];  // [NI, 64]
  const float* W1   = (const float*)d_in[4];
  const float* b1   = (const float*)d_in[5];
  const float* W2   = (const float*)d_in[6];
  const float* b2   = (const float*)d_in[7];

  long long E = in_sizes[1];
  int NU = in_sizes[2] / D;
  int NI = in_sizes[3] / D;
  int N  = NU + NI;

  const int* srcI = eidx;
  const int* dstI = eidx + E;

  float* agg  = (float*)d_ws;
  float* x1   = agg + (long long)N * D;
  float* outp = (float*)d_out;

  long long n4    = (long long)N * D / 4;
  int       zb    = (int)((n4 + 255) / 256);
  long long nwork = E * 16;
  int       eb    = (int)((nwork + 255) / 256);
  int       nt    = (N + 15) / 16;
  int       gb    = (nt + 7) / 8;

  // ---- Layer 1 ----
  SENGR_zero_f32<<<zb, 256, 0, stream>>>(agg, n4);
  SENGR_edge_scatter<<<eb, 256, 0, stream>>>(srcI, dstI, ew, ue, ie, NU, agg,
                                             nwork);
  SENGR_gcn_linear_wmma<<<gb, 256, 0, stream>>>(agg, ue, ie, NU, W1, b1, x1, N);

  // ---- Layer 2 ----
  SENGR_zero_f32<<<zb, 256, 0, stream>>>(agg, n4);
  SENGR_edge_scatter<<<eb, 256, 0, stream>>>(srcI, dstI, ew, x1, x1, N, agg,
                                             nwork);
  SENGR_gcn_linear_wmma<<<gb, 256, 0, stream>>>(agg, x1, x1, N, W2, b2, outp,
                                                N);
}